// PlaneValidator_24635932410264
// MI455X (gfx1250) — compile-verified
//
#include <hip/hip_runtime.h>
#include <math.h>

typedef __attribute__((ext_vector_type(2))) float v2f;
typedef __attribute__((ext_vector_type(8))) float v8f;

#define COS_THR 0.86602540378443864676f   // cos(pi/6)
#define FLT_BIG 3.402823466e+38f

// order-preserving float <-> uint map so ds_min_u32 implements float min
__device__ __forceinline__ unsigned flipF(float f) {
  unsigned u = __float_as_uint(f);
  return (u & 0x80000000u) ? ~u : (u | 0x80000000u);
}
__device__ __forceinline__ float unflipF(unsigned u) {
  return (u & 0x80000000u) ? __uint_as_float(u & 0x7FFFFFFFu)
                           : __uint_as_float(~u);
}

// One block per (b,h). D[n,m] = P2[n] + R2[m] - 2 p_n . r_m generated tile-by-tile
// with V_WMMA_F32_16X16X4_F32 (A=[p,1], B=[-2r,R2], C preloaded with P2).
__global__ __launch_bounds__(256)
void sde_chamfer_wmma(const float* __restrict__ sp,   // (4,1024,3)
                      const float* __restrict__ yp,   // (4,16,4)
                      float* __restrict__ sdeOut)     // (64,)
{
  // fragment-interleaved LDS: one ds_load_b64 yields a whole per-lane fragment
  __shared__ float2 sA[2][1024];   // [0][j]=(px,py)  [1][j]=(pz,1)     (K01 / K23)
  __shared__ float2 sB[2][1024];   // [0][j]=(-2rx,-2ry) [1][j]=(-2rz,R2)
  __shared__ float  sP2[1024], sRow[1024];
  __shared__ unsigned sColU[1024];
  __shared__ float  sPartial[256];

  const int bid  = blockIdx.x;
  const int b    = bid >> 4, h = bid & 15;
  const int tid  = threadIdx.x;
  const int lane = tid & 31, wave = tid >> 5;

  // plane params (wave-uniform)
  const float nx = yp[(b*16+h)*4+0], ny = yp[(b*16+h)*4+1],
              nz = yp[(b*16+h)*4+2], dd = yp[(b*16+h)*4+3];
  const float inv = 1.0f / sqrtf(nx*nx + ny*ny + nz*nz);
  const float nhx = nx*inv, nhy = ny*inv, nhz = nz*inv, dh = dd*inv;

  // stage points + reflections in LDS
  const float* P = sp + b*1024*3;
  for (int k = 0; k < 4; ++k) {
    int j = tid + k*256;
    float px = P[j*3+0], py = P[j*3+1], pz = P[j*3+2];
    float p2   = px*px + py*py + pz*pz;
    float proj = px*nhx + py*nhy + pz*nhz + dh;
    float rx = px - 2.0f*proj*nhx;
    float ry = py - 2.0f*proj*nhy;
    float rz = pz - 2.0f*proj*nhz;
    sA[0][j] = make_float2(px, py);
    sA[1][j] = make_float2(pz, 1.0f);
    sB[0][j] = make_float2(-2.0f*rx, -2.0f*ry);
    sB[1][j] = make_float2(-2.0f*rz, rx*rx + ry*ry + rz*rz);
    sP2[j]   = p2;
    sColU[j] = 0xFFFFFFFFu;
  }
  __syncthreads();

  const int  half = (lane >> 4);          // 0: K=0,1 side   1: K=2,3 side
  const int  l15  = lane & 15;
  const float2* aPtr = sA[half];
  const float2* bPtr = sB[half];

  // each wave owns 8 row tiles (exclusive rows -> no atomics for row mins)
  for (int rti = 0; rti < 8; ++rti) {
    const int rt = wave + rti*8;
    const int rbase = rt*16;

    // A fragment (16x4 f32), one b64 load per lane
    float2 av = aPtr[rbase + l15];
    v2f a; a[0] = av.x; a[1] = av.y;

    // C init: P2 broadcast along rows (VGPR i -> M=i / M=i+8)
    v8f cini;
    #pragma unroll
    for (int i = 0; i < 8; ++i)
      cini[i] = sP2[rbase + i + half*8];

    float rowAcc[8];
    #pragma unroll
    for (int i = 0; i < 8; ++i) rowAcc[i] = FLT_BIG;

    // double-buffered column sweep: prefetch B for ct+1 while doing WMMA for ct
    float2 bv = bPtr[l15];
    for (int ct = 0; ct < 64; ++ct) {
      float2 bnext = bPtr[(((ct + 1) & 63) << 4) + l15];
      v2f bf; bf[0] = bv.x; bf[1] = bv.y;

      // D tile = A*B + C
      v8f d = __builtin_amdgcn_wmma_f32_16x16x4_f32(
          false, a, false, bf, (short)0, cini, false, false);

      float cmin = fminf(d[0], d[1]);
      #pragma unroll
      for (int i = 0; i < 8; ++i) rowAcc[i] = fminf(rowAcc[i], d[i]);
      #pragma unroll
      for (int i = 2; i < 8; ++i) cmin = fminf(cmin, d[i]);

      // all 32 lanes atomic-min; lanes l and l+16 share an address, so the
      // LDS atomic unit performs the half-wave combine (no bpermute, no branch)
      atomicMin(&sColU[(ct << 4) + l15], flipF(cmin));

      bv = bnext;
    }

    // row min: reduce across the 16 lanes of each half
    #pragma unroll
    for (int i = 0; i < 8; ++i) {
      float v = rowAcc[i];
      v = fminf(v, __shfl_xor(v, 1, 32));
      v = fminf(v, __shfl_xor(v, 2, 32));
      v = fminf(v, __shfl_xor(v, 4, 32));
      v = fminf(v, __shfl_xor(v, 8, 32));
      if (l15 == 0) sRow[rbase + i + half*8] = v;
    }
  }
  __syncthreads();

  // sde = mean(rowmin) + mean(colmin)
  float part = 0.0f;
  for (int k = 0; k < 4; ++k) {
    int j = tid*4 + k;
    part += sRow[j] + unflipF(sColU[j]);
  }
  sPartial[tid] = part;
  __syncthreads();
  if (tid == 0) {
    float tot = 0.0f;
    for (int t = 0; t < 256; ++t) tot += sPartial[t];
    sdeOut[bid] = tot * (1.0f/1024.0f);
  }
}

// Per-batch: conf, stable sort by -conf, cosine NMS, compact kept rows.
__global__ void plane_nms(const float* __restrict__ yp,
                          const float* __restrict__ sde,
                          const int* __restrict__ thrI,
                          float* __restrict__ out)     // (4,16,8)
{
  if (threadIdx.x != 0) return;
  const int b = blockIdx.x;
  float nh[16][3], pt[16][3], cf[16], sd[16];
  float mn = FLT_BIG, mx = -FLT_BIG;
  for (int h = 0; h < 16; ++h) {
    const float* y = yp + (b*16+h)*4;
    float inv = 1.0f / sqrtf(y[0]*y[0]+y[1]*y[1]+y[2]*y[2]);
    nh[h][0]=y[0]*inv; nh[h][1]=y[1]*inv; nh[h][2]=y[2]*inv;
    float dh = y[3]*inv;
    pt[h][0]=-dh*nh[h][0]; pt[h][1]=-dh*nh[h][1]; pt[h][2]=-dh*nh[h][2];
    sd[h] = sde[b*16+h];
    mn = fminf(mn, sd[h]); mx = fmaxf(mx, sd[h]);
  }
  const float scl = 1.0f/(mx-mn);
  for (int h = 0; h < 16; ++h) cf[h] = 1.0f - (sd[h]-mn)*scl;
  const float thr = (float)thrI[0];

  // stable argsort(-conf)
  int order[16];
  for (int i = 0; i < 16; ++i) {
    int r = 0;
    for (int j = 0; j < 16; ++j)
      if (cf[j] > cf[i] || (cf[j] == cf[i] && j < i)) ++r;
    order[r] = i;
  }
  bool keep[16];
  for (int k = 0; k < 16; ++k) keep[k] = (sd[order[k]] <= thr);
  for (int i = 0; i < 16; ++i) {
    if (!keep[i]) continue;
    for (int j = i+1; j < 16; ++j) {
      if (!keep[j]) continue;
      float c = nh[order[i]][0]*nh[order[j]][0]
              + nh[order[i]][1]*nh[order[j]][1]
              + nh[order[i]][2]*nh[order[j]][2];
      c = fminf(1.0f, fmaxf(-1.0f, c));
      if (c > COS_THR) keep[j] = false;
    }
  }
  // stable partition: kept rows first (in conf order), zeros after
  float* ob = out + b*16*8;
  int slot = 0;
  for (int k = 0; k < 16; ++k) {
    if (!keep[k]) continue;
    int i = order[k];
    ob[slot*8+0]=nh[i][0]; ob[slot*8+1]=nh[i][1]; ob[slot*8+2]=nh[i][2];
    ob[slot*8+3]=pt[i][0]; ob[slot*8+4]=pt[i][1]; ob[slot*8+5]=pt[i][2];
    ob[slot*8+6]=cf[i];    ob[slot*8+7]=sd[i];
    ++slot;
  }
  for (; slot < 16; ++slot)
    for (int c = 0; c < 8; ++c) ob[slot*8+c] = 0.0f;
}

extern "C" void kernel_launch(void* const* d_in, const int* in_sizes, int n_in,
                              void* d_out, int out_size, void* d_ws, size_t ws_size,
                              hipStream_t stream) {
  (void)in_sizes; (void)n_in; (void)out_size; (void)ws_size;
  const float* sp  = (const float*)d_in[0];   // sample_points (4,1024,3)
  const float* yp  = (const float*)d_in[1];   // y_pred (4,16,4)
  const int*   thr = (const int*)d_in[2];     // sde_threshold
  float* out = (float*)d_out;                 // (4,16,8)
  float* sde = (float*)d_ws;                  // 64 floats scratch

  sde_chamfer_wmma<<<64, 256, 0, stream>>>(sp, yp, sde);
  plane_nms<<<4, 32, 0, stream>>>(yp, sde, thr, out);
}